// PWTDConv1d_70257075028074
// MI455X (gfx1250) — compile-verified
//
#include <hip/hip_runtime.h>

typedef float v2f __attribute__((ext_vector_type(2)));
typedef float v8f __attribute__((ext_vector_type(8)));

#define BATCH 8
#define CIN   512
#define TLEN  512
#define OCH   512
#define KSZ   65
#define PADW  32

#define NT 4           // t-tiles computed per wave (amortizes B-band + offset setup)
#define KT 20          // K-steps of 4: band of 80 channels per 16-wide o-tile

// y[b,t,o] = sum_{c=o-32}^{o+32} x[b,c,t] * w[o, c-o+32]  (zero padded in c)
// Banded GEMM via V_WMMA_F32_16X16X4_F32 (full fp32 precision).
// All masking (kernel band + x zero-pad) is folded into the B fragments at
// wave init, so the hot loop is pure {global_load_b32 x2, v_wmma} per K-step.
__global__ __launch_bounds__(256) void pwtdconv_wmma_f32(
    const float* __restrict__ x,     // (B, CIN, TLEN)
    const float* __restrict__ w,     // (OCH, 1, KSZ)
    const float* __restrict__ bias,  // (OCH, 1)
    float* __restrict__ out)         // (B, TLEN, OCH)
{
    const int gwave = (int)((blockIdx.x * blockDim.x + threadIdx.x) >> 5);
    const int lane  = (int)(threadIdx.x & 31);
    const int lm    = lane & 15;   // M (A row) / N (B,D col) within tile
    const int lh    = lane >> 4;   // selects K pair {0,1} vs {2,3}; D row +8

    const int OT = OCH / 16;           // 32 o-tiles
    const int TG = (TLEN / 16) / NT;   // 8 t-groups

    int idx = gwave;
    const int ot = idx % OT; idx /= OT;
    const int tg = idx % TG; idx /= TG;
    const int b  = idx;
    if (b >= BATCH) return;

    const int o0    = ot * 16;
    const int o     = o0 + lm;         // this lane's output channel (B col / D col)
    const int cbase = o0 - PADW;       // first channel of the 80-wide band window

    // ---- Wave-invariant setup: banded-B fragments + A byte-offsets ----
    // B[K][N] layout: VGPR r, lane -> K = r + 2*lh, N = lm
    // A[M][K] layout: VGPR r, lane -> M = lm,       K = r + 2*lh
    v2f      bfrag[KT];
    unsigned aoff[KT][2];
#pragma unroll
    for (int kt = 0; kt < KT; ++kt) {
#pragma unroll
        for (int r = 0; r < 2; ++r) {
            const int c  = cbase + kt * 4 + 2 * lh + r;
            const int kk = c - o + PADW;                       // tap index in 65-wide kernel
            const bool inb = (kk >= 0) & (kk < KSZ) & (c >= 0) & (c < CIN);
            const int  kc  = kk < 0 ? 0 : (kk > KSZ - 1 ? KSZ - 1 : kk);
            const float wv = w[o * KSZ + kc];
            bfrag[kt][r] = inb ? wv : 0.0f;   // band mask AND x zero-pad folded into B

            const int cc = c < 0 ? 0 : (c > CIN - 1 ? CIN - 1 : c);  // always-valid address
            aoff[kt][r]  = (unsigned)((cc * TLEN + lm) * 4);         // byte offset in x[b]
        }
    }

    const float bv = bias[o];
    const char* xb = (const char*)(x + (size_t)b * CIN * TLEN);
    float*      ob = out + (size_t)b * TLEN * OCH;

    for (int it = 0; it < NT; ++it) {
        const int   t0  = (tg * NT + it) * 16;      // wave-uniform
        const char* xbt = xb + (size_t)t0 * 4;      // scalar base bump per t-tile

        v8f acc;
#pragma unroll
        for (int r = 0; r < 8; ++r) acc[r] = bv;    // bias as the initial C matrix

#pragma unroll
        for (int kt = 0; kt < KT; ++kt) {
            v2f a;
            a[0] = *(const float*)(xbt + aoff[kt][0]);
            a[1] = *(const float*)(xbt + aoff[kt][1]);
            // D = A(16x4,f32) * B(4x16,f32) + C(16x16,f32)
            acc = __builtin_amdgcn_wmma_f32_16x16x4_f32(
                false, a, false, bfrag[kt], (short)0, acc, false, false);
        }

        // D layout: VGPR r, lanes 0-15 -> row t0+r; lanes 16-31 -> row t0+r+8; col = o
        float* obt = ob + (size_t)(t0 + 8 * lh) * OCH + o;
#pragma unroll
        for (int r = 0; r < 8; ++r)
            obt[(size_t)r * OCH] = acc[r];
    }
}

extern "C" void kernel_launch(void* const* d_in, const int* in_sizes, int n_in,
                              void* d_out, int out_size, void* d_ws, size_t ws_size,
                              hipStream_t stream) {
    const float* x    = (const float*)d_in[0];   // 8*512*512
    const float* w    = (const float*)d_in[1];   // 512*1*65
    const float* bias = (const float*)d_in[2];   // 512*1
    float*       out  = (float*)d_out;           // 8*512*512

    // total waves = BATCH * (OCH/16) * (TLEN/16/NT) = 8*32*8 = 2048 -> 256 blocks of 8 waves
    const int total_waves   = BATCH * (OCH / 16) * ((TLEN / 16) / NT);
    const int block_threads = 256;
    const int grid_blocks   = (total_waves * 32 + block_threads - 1) / block_threads;

    hipLaunchKernelGGL(pwtdconv_wmma_f32, dim3(grid_blocks), dim3(block_threads),
                       0, stream, x, w, bias, out);
}